// TilePositionEmbedding_15848429323035
// MI455X (gfx1250) — compile-verified
//
#include <hip/hip_runtime.h>
#include <hip/hip_bf16.h>
#include <cmath>

// Problem constants (from reference setup_inputs).
#define B_  16
#define T_  4
#define N_  1601
#define D_  1280
#define D4  (D_ / 4)                 // 320 float4 per PE row == blockDim.x
#define ROWS_PER_BLOCK 10
#define NBLK_X ((N_ + ROWS_PER_BLOCK - 1) / ROWS_PER_BLOCK)   // 161

typedef float __attribute__((ext_vector_type(4))) fv4;
typedef int   __attribute__((ext_vector_type(4))) iv4;
typedef __attribute__((address_space(1))) iv4 glob_iv4;   // __device__ int4
typedef __attribute__((address_space(3))) iv4 lds_iv4;    // __shared__ int4

#if defined(__has_builtin)
#  if __has_builtin(__builtin_amdgcn_global_load_async_to_lds_b128)
#    define HAVE_ASYNC_BUILTIN 1
#  endif
#  if __has_builtin(__builtin_amdgcn_s_wait_asynccnt)
#    define HAVE_WAIT_ASYNC_BUILTIN 1
#  endif
#endif

__global__ __launch_bounds__(D4) void tile_pe_add_kernel(
    const fv4* __restrict__ x,          // (B,T,N,D) as float4
    const int* __restrict__ aspect,     // (B,2)
    const float* __restrict__ emb,      // (T,T,1,D)
    const float* __restrict__ gate,     // (1,)
    fv4* __restrict__ out)              // (B,T,N,D) as float4
{
    __shared__ fv4 pe_lds[D4];          // one PE row: 1280 floats = 5 KB

    const int tid = threadIdx.x;        // 0..319, == float4 index inside D
    const int bt  = blockIdx.y;         // 0..63
    const int b   = bt >> 2;
    const int t   = bt & 3;

    const int  h     = aspect[2 * b];
    const int  w     = aspect[2 * b + 1];
    const bool valid = t < h * w;
    const int  row   = valid ? (t / w) : 0;
    const int  col   = valid ? (t % w) : 0;
    const float g    = valid ? tanhf(gate[0]) : 0.0f;   // invalid -> pe*0 == reference's where()

    const fv4* src = (const fv4*)emb + (size_t)(row * T_ + col) * D4;

    // --- Stage the PE row into LDS via the CDNA5 async-copy path (ASYNCcnt). ---
#if HAVE_ASYNC_BUILTIN
    __builtin_amdgcn_global_load_async_to_lds_b128(
        (glob_iv4*)(src + tid),
        (lds_iv4*)&pe_lds[tid],
        /*imm offset*/ 0, /*cpol*/ 0);
#else
    {
        // LDS flat address: low 32 bits of the generic pointer are the LDS byte offset.
        unsigned   lds_b = (unsigned)(unsigned long long)(void*)&pe_lds[tid];
        const fv4* gptr  = src + tid;
        asm volatile("global_load_async_to_lds_b128 %0, %1, off"
                     :: "v"(lds_b), "v"(gptr) : "memory");
    }
#endif
#if HAVE_WAIT_ASYNC_BUILTIN
    __builtin_amdgcn_s_wait_asynccnt(0);
#else
    asm volatile("s_wait_asynccnt 0" ::: "memory");
#endif
    __syncthreads();

    // Each thread owns one float4 of the (already gated) PE row in registers.
    fv4 pv = pe_lds[tid];
    pv *= g;

    // --- Stream 10 rows of x per block: non-temporal b128 load/store. ---
    const int    n0   = blockIdx.x * ROWS_PER_BLOCK;
    const size_t base = ((size_t)bt * N_ + (size_t)n0) * D4 + (size_t)tid;

#pragma unroll
    for (int r = 0; r < ROWS_PER_BLOCK; ++r) {
        if (n0 + r < N_) {
            const size_t off = base + (size_t)r * D4;
            fv4 v = __builtin_nontemporal_load(&x[off]);
            v += pv;
            __builtin_nontemporal_store(v, &out[off]);
        }
    }
}

extern "C" void kernel_launch(void* const* d_in, const int* in_sizes, int n_in,
                              void* d_out, int out_size, void* d_ws, size_t ws_size,
                              hipStream_t stream) {
    (void)in_sizes; (void)n_in; (void)out_size; (void)d_ws; (void)ws_size;

    const fv4*   x      = (const fv4*)d_in[0];
    const int*   aspect = (const int*)d_in[1];
    const float* emb    = (const float*)d_in[2];
    const float* gate   = (const float*)d_in[3];
    fv4*         out    = (fv4*)d_out;

    dim3 grid(NBLK_X, B_ * T_, 1);   // (161, 64)
    dim3 block(D4, 1, 1);            // 320 threads = 10 waves
    tile_pe_add_kernel<<<grid, block, 0, stream>>>(x, aspect, emb, gate, out);
}